// subcluster_ddfm_loss_29652454211787
// MI455X (gfx1250) — compile-verified
//
#include <hip/hip_runtime.h>

typedef float v2f __attribute__((ext_vector_type(2)));
typedef float v8f __attribute__((ext_vector_type(8)));
typedef int   v4i __attribute__((ext_vector_type(4)));

#define DM 128        // feat_dim (reference fixed)
#define SUBC 3        // NUM_SUBCLUSTER
#define MARGINF 1.0f
#define CSTRIP 4      // column tiles per wave (16x64 output strip)
#define BWAVES 4      // waves per block in the WMMA kernels
#define LROW (DM + 4) // padded LDS row stride (floats) -> conflict-free b64 fragment reads
#define WAVE_LDS_ROWS (16 + 16 * CSTRIP)              // 80 rows (A tile + 4 B tiles)
#define WAVE_LDS_FLOATS (WAVE_LDS_ROWS * LROW)        // 80 * 132 = 10560
#define BLOCK_LDS_BYTES (BWAVES * WAVE_LDS_FLOATS * 4) // 168960 B <= 320KB/WGP

// ---- CDNA5 async global->LDS copy (ASYNCcnt-tracked, no VGPR landing zone) ----
__device__ __forceinline__ void async_row16(const float* g, float* l) {
    // per-lane: copy 16 bytes; whole wave copies one 512B row
#if __has_builtin(__builtin_amdgcn_global_load_async_to_lds_b128)
    __builtin_amdgcn_global_load_async_to_lds_b128(
        (__attribute__((address_space(1))) v4i*)g,
        (__attribute__((address_space(3))) v4i*)l, 0, 0);
#else
    unsigned lofs = (unsigned)(unsigned long long)(__attribute__((address_space(3))) char*)(void*)l;
    asm volatile("global_load_async_to_lds_b128 %0, %1, off"
                 :: "v"(lofs), "v"(g) : "memory");
#endif
}

__device__ __forceinline__ void async_wait0() {
#if __has_builtin(__builtin_amdgcn_s_wait_asynccnt)
    __builtin_amdgcn_s_wait_asynccnt(0);
#else
    asm volatile("s_wait_asynccnt 0x0" ::: "memory");
#endif
    asm volatile("" ::: "memory");  // compiler barrier: LDS reads stay after the wait
}

// ---------------- K0: c2[n], present[n]=0, zero the 3 outputs ----------------
__global__ void k_setup(const float* __restrict__ centers, float* __restrict__ c2,
                        int* __restrict__ present, float* __restrict__ out, int N) {
    int n = blockIdx.x * blockDim.x + threadIdx.x;
    if (n < 3) out[n] = 0.0f;
    if (n >= N) return;
    const float* c = centers + (size_t)n * DM;
    float s = 0.0f;
#pragma unroll 8
    for (int d = 0; d < DM; ++d) s += c[d] * c[d];
    c2[n] = s;
    present[n] = 0;
}

// ---------------- K1: scatter presence flags --------------------------------
__global__ void k_scatter(const int* __restrict__ labels, int* __restrict__ present, int B) {
    int b = blockIdx.x * blockDim.x + threadIdx.x;
    if (b < B) present[labels[b]] = 1;
}

// ---------------- K2: x2[b], intra_d[b], intraclass loss --------------------
__global__ void k_intra(const float* __restrict__ x, const float* __restrict__ centers,
                        const int* __restrict__ labels, const float* __restrict__ c2,
                        float* __restrict__ x2v, float* __restrict__ intra,
                        float* __restrict__ out, int B, float scale) {
    int b = blockIdx.x * blockDim.x + threadIdx.x;
    if (b >= B) return;
    const float* xr = x + (size_t)b * DM;
    int lab = labels[b];
    const float* cr = centers + (size_t)lab * DM;
    float sx = 0.0f, sd = 0.0f;
#pragma unroll 8
    for (int d = 0; d < DM; ++d) { float xv = xr[d]; sx += xv * xv; sd += xv * cr[d]; }
    x2v[b] = sx;
    float id = sx + c2[lab] - 2.0f * sd;
    intra[b] = id;
    atomicAdd(out + 0, id * scale);
}

// ---------------- K3: per-class max pairwise dist -> rowmax/rowvalid --------
__global__ void k_classmax(const float* __restrict__ centers, const int* __restrict__ present,
                           const float* __restrict__ c2, float* __restrict__ rowmax,
                           int* __restrict__ rowvalid, int C) {
    int c = blockIdx.x * blockDim.x + threadIdx.x;
    if (c >= C) return;
    int base = c * SUBC;
    int p0 = present[base], p1 = present[base + 1], p2 = present[base + 2];
    int cnt = p0 + p1 + p2;
    int inc = cnt > 1;
    float cm = 0.0f;
    if (inc) {
        const float* a0 = centers + (size_t)base * DM;
        const float* a1 = a0 + DM;
        const float* a2 = a1 + DM;
        float d01 = 0.f, d02 = 0.f, d12 = 0.f;
#pragma unroll 8
        for (int d = 0; d < DM; ++d) {
            float v0 = a0[d], v1 = a1[d], v2 = a2[d];
            d01 += v0 * v1; d02 += v0 * v2; d12 += v1 * v2;
        }
        float q0 = c2[base], q1 = c2[base + 1], q2 = c2[base + 2];
        float m = 0.0f;  // diagonal Dcc[i,i]~0 is always in the reference max
        if (p0 && p1) m = fmaxf(m, q0 + q1 - 2.0f * d01);
        if (p0 && p2) m = fmaxf(m, q0 + q2 - 2.0f * d02);
        if (p1 && p2) m = fmaxf(m, q1 + q2 - 2.0f * d12);
        cm = m;
    }
#pragma unroll
    for (int s = 0; s < SUBC; ++s) {
        rowmax[base + s] = cm;
        rowvalid[base + s] = present[base + s] && inc;
    }
}

// ---------------- K4: async-LDS staged WMMA x @ centers^T + triplet hinge ---
__global__ __launch_bounds__(BWAVES * 32) void k_triplet(
    const float* __restrict__ x, const float* __restrict__ centers,
    const int* __restrict__ labels, const float* __restrict__ c2,
    const float* __restrict__ x2v, const float* __restrict__ intra,
    const int* __restrict__ present, float* __restrict__ out,
    int B, int N, float scale) {
    extern __shared__ float smem[];
    int lane = threadIdx.x & 31;
    int wave = threadIdx.x >> 5;
    int tiles_n = N >> 4;
    int strips = (tiles_n + CSTRIP - 1) / CSTRIP;
    long w = (long)blockIdx.x * BWAVES + wave;
    long nwork = (long)(B >> 4) * strips;
    if (w >= nwork) return;
    int tr = (int)(w / strips), sc = (int)(w % strips);
    int row0 = tr << 4;
    int half = lane >> 4;   // K sub-block per ISA A/B layout
    int lid = lane & 15;

    int col0[CSTRIP];
    int valid[CSTRIP];
#pragma unroll
    for (int t = 0; t < CSTRIP; ++t) {
        int ct = sc * CSTRIP + t;
        valid[t] = ct < tiles_n;
        int ctc = valid[t] ? ct : (tiles_n - 1);   // clamp: in-bounds duplicate work
        col0[t] = ctc << 4;
    }

    float* aw = smem + wave * WAVE_LDS_FLOATS;   // 16 rows of x
    float* bw = aw + 16 * LROW;                  // 4 * 16 rows of centers

    // ---- stage the whole K-slab asynchronously into LDS ----
#pragma unroll
    for (int r = 0; r < 16; ++r)
        async_row16(x + (size_t)(row0 + r) * DM + lane * 4, aw + r * LROW + lane * 4);
#pragma unroll
    for (int t = 0; t < CSTRIP; ++t)
#pragma unroll
        for (int r = 0; r < 16; ++r)
            async_row16(centers + (size_t)(col0[t] + r) * DM + lane * 4,
                        bw + (t * 16 + r) * LROW + lane * 4);
    async_wait0();

    // ---- GEMM from LDS (bank-conflict-free padded rows) ----
    v8f acc[CSTRIP];
#pragma unroll
    for (int t = 0; t < CSTRIP; ++t) acc[t] = (v8f){0.f,0.f,0.f,0.f,0.f,0.f,0.f,0.f};

    const float* afrag = aw + lid * LROW + half * 2;
#pragma unroll
    for (int kk = 0; kk < DM / 4; ++kk) {
        v2f a = *(const v2f*)(afrag + kk * 4);
#pragma unroll
        for (int t = 0; t < CSTRIP; ++t) {
            v2f b = *(const v2f*)(bw + (t * 16 + lid) * LROW + half * 2 + kk * 4);
            acc[t] = __builtin_amdgcn_wmma_f32_16x16x4_f32(false, a, false, b,
                                                           (short)0, acc[t], false, false);
        }
    }

    // column-side scalars (one per tile)
    float cn2[CSTRIP]; int pn[CSTRIP]; int ncls[CSTRIP];
#pragma unroll
    for (int t = 0; t < CSTRIP; ++t) {
        int n = col0[t] + lid;
        cn2[t] = c2[n];
        pn[t] = present[n];
        ncls[t] = n / SUBC;
    }

    float sum = 0.0f;
#pragma unroll
    for (int j = 0; j < 8; ++j) {
        int row = row0 + j + half * 8;               // C/D layout: VGPR j -> M=j (+8 hi half)
        float xb2 = x2v[row];
        float idr = intra[row];
        int lcls = labels[row] / SUBC;
#pragma unroll
        for (int t = 0; t < CSTRIP; ++t) {
            float dxc = xb2 + cn2[t] - 2.0f * acc[t][j];
            float v = fmaxf(MARGINF + (idr - dxc), 0.0f);
            bool keep = valid[t] && !(pn[t] && (lcls == ncls[t]));
            sum += keep ? v : 0.0f;
        }
    }
#pragma unroll
    for (int off = 16; off; off >>= 1) sum += __shfl_xor(sum, off, 32);
    if (lane == 0) atomicAdd(out + 2, sum * scale);
}

// ---------------- K5: async-LDS staged WMMA centers @ centers^T + hinge -----
__global__ __launch_bounds__(BWAVES * 32) void k_inter(
    const float* __restrict__ centers, const float* __restrict__ c2,
    const int* __restrict__ present, const float* __restrict__ rowmax,
    const int* __restrict__ rowvalid, float* __restrict__ out,
    int N, float scale) {
    extern __shared__ float smem[];
    int lane = threadIdx.x & 31;
    int wave = threadIdx.x >> 5;
    int tiles_n = N >> 4;
    int strips = (tiles_n + CSTRIP - 1) / CSTRIP;
    long w = (long)blockIdx.x * BWAVES + wave;
    long nwork = (long)tiles_n * strips;
    if (w >= nwork) return;
    int tr = (int)(w / strips), sc = (int)(w % strips);
    int row0 = tr << 4;
    int half = lane >> 4;
    int lid = lane & 15;

    // Early-out: skip the whole strip if all 16 rows are invalid (~half of rows).
    // Ballot result is wave-uniform, so EXEC stays all-1s for the WMMA path.
    int rv16 = (lane < 16) ? rowvalid[row0 + lane] : 0;
    if (__ballot(rv16 != 0) == 0ull) return;

    int col0[CSTRIP];
    int valid[CSTRIP];
#pragma unroll
    for (int t = 0; t < CSTRIP; ++t) {
        int ct = sc * CSTRIP + t;
        valid[t] = ct < tiles_n;
        int ctc = valid[t] ? ct : (tiles_n - 1);
        col0[t] = ctc << 4;
    }

    float* aw = smem + wave * WAVE_LDS_FLOATS;
    float* bw = aw + 16 * LROW;

#pragma unroll
    for (int r = 0; r < 16; ++r)
        async_row16(centers + (size_t)(row0 + r) * DM + lane * 4, aw + r * LROW + lane * 4);
#pragma unroll
    for (int t = 0; t < CSTRIP; ++t)
#pragma unroll
        for (int r = 0; r < 16; ++r)
            async_row16(centers + (size_t)(col0[t] + r) * DM + lane * 4,
                        bw + (t * 16 + r) * LROW + lane * 4);
    async_wait0();

    v8f acc[CSTRIP];
#pragma unroll
    for (int t = 0; t < CSTRIP; ++t) acc[t] = (v8f){0.f,0.f,0.f,0.f,0.f,0.f,0.f,0.f};

    const float* afrag = aw + lid * LROW + half * 2;
#pragma unroll
    for (int kk = 0; kk < DM / 4; ++kk) {
        v2f a = *(const v2f*)(afrag + kk * 4);
#pragma unroll
        for (int t = 0; t < CSTRIP; ++t) {
            v2f b = *(const v2f*)(bw + (t * 16 + lid) * LROW + half * 2 + kk * 4);
            acc[t] = __builtin_amdgcn_wmma_f32_16x16x4_f32(false, a, false, b,
                                                           (short)0, acc[t], false, false);
        }
    }

    float cj2[CSTRIP]; int pj[CSTRIP]; int jcls[CSTRIP];
#pragma unroll
    for (int t = 0; t < CSTRIP; ++t) {
        int jc = col0[t] + lid;
        cj2[t] = c2[jc];
        pj[t] = present[jc];
        jcls[t] = jc / SUBC;
    }

    float sum = 0.0f;
#pragma unroll
    for (int r = 0; r < 8; ++r) {
        int i = row0 + r + half * 8;
        float ci2 = c2[i];
        float rm = rowmax[i];
        int rv = rowvalid[i];
        int icls = i / SUBC;
#pragma unroll
        for (int t = 0; t < CSTRIP; ++t) {
            float dcc = ci2 + cj2[t] - 2.0f * acc[t][r];
            float v = fmaxf(MARGINF + rm - dcc, 0.0f);
            bool keep = valid[t] && rv && !(pj[t] && (icls == jcls[t]));
            sum += keep ? v : 0.0f;
        }
    }
#pragma unroll
    for (int off = 16; off; off >>= 1) sum += __shfl_xor(sum, off, 32);
    if (lane == 0) atomicAdd(out + 1, sum * scale);
}

extern "C" void kernel_launch(void* const* d_in, const int* in_sizes, int n_in,
                              void* d_out, int out_size, void* d_ws, size_t ws_size,
                              hipStream_t stream) {
    (void)n_in; (void)out_size; (void)ws_size;
    const float* x       = (const float*)d_in[0];
    const float* centers = (const float*)d_in[1];
    const int*   labels  = (const int*)d_in[2];
    float* out = (float*)d_out;

    int B = in_sizes[2];            // 4096
    int D = in_sizes[0] / B;        // 128 (== DM)
    int N = in_sizes[1] / D;        // 6000
    int C = N / SUBC;               // 2000
    (void)D;

    // workspace layout (floats): c2[N] | rowmax[N] | x2[B] | intra[B] | present[N] | rowvalid[N]
    float* ws       = (float*)d_ws;
    float* c2       = ws;
    float* rowmax   = c2 + N;
    float* x2v      = rowmax + N;
    float* intra    = x2v + B;
    int*   present  = (int*)(intra + B);
    int*   rowvalid = present + N;

    k_setup<<<(N + 255) / 256, 256, 0, stream>>>(centers, c2, present, out, N);
    k_scatter<<<(B + 255) / 256, 256, 0, stream>>>(labels, present, B);

    float s_intra = 1.0f / ((float)B * (float)DM * 2.0f);
    k_intra<<<(B + 255) / 256, 256, 0, stream>>>(x, centers, labels, c2, x2v, intra, out, B, s_intra);

    k_classmax<<<(C + 255) / 256, 256, 0, stream>>>(centers, present, c2, rowmax, rowvalid, C);

    int tiles_n = N >> 4;
    int strips = (tiles_n + CSTRIP - 1) / CSTRIP;

    long t_work = (long)(B >> 4) * strips;
    float s_trip = 1.0f / (2.0f * (float)N * (float)B);
    k_triplet<<<(unsigned)((t_work + BWAVES - 1) / BWAVES), BWAVES * 32, BLOCK_LDS_BYTES, stream>>>(
        x, centers, labels, c2, x2v, intra, present, out, B, N, s_trip);

    long i_work = (long)tiles_n * strips;
    float s_inter = 1.0f / ((float)N * (float)B * 2.0f);
    k_inter<<<(unsigned)((i_work + BWAVES - 1) / BWAVES), BWAVES * 32, BLOCK_LDS_BYTES, stream>>>(
        centers, c2, present, rowmax, rowvalid, out, N, s_inter);
}